// CrossScanMamba_56281251446876
// MI455X (gfx1250) — compile-verified
//
#include <hip/hip_runtime.h>
#include <hip/hip_bf16.h>
#include <math.h>

// ---------------- problem constants ----------------
#define D_MODEL 128
#define DI      512          // EXPAND * D
#define NST     64           // N_STATE
#define DTR     8            // DT_RANK
#define NXP     192          // x_dbl width padded 136 -> 192 (multiple of 64)
#define BATCH   8
#define LSEG    512
#define LTOT    1536         // Lt+La+Lv
#define TTOK    (BATCH*LTOT) // 12288 token rows
#define QSP     8            // states per scan thread (64/8 octets per channel)

typedef float v2f __attribute__((ext_vector_type(2)));
typedef float v8f __attribute__((ext_vector_type(8)));

// ---------------- concat z_t,z_a,z_v -> x_seq (T x 128) ----------------
__global__ void k_concat(const float* __restrict__ zt, const float* __restrict__ za,
                         const float* __restrict__ zv, float* __restrict__ xs) {
  int i = blockIdx.x * blockDim.x + threadIdx.x;
  if (i >= TTOK * D_MODEL) return;
  int r = i / D_MODEL, c = i % D_MODEL;
  int b = r / LTOT, l = r % LTOT;
  const float* src; int rr;
  if (l < LSEG)            { src = zt; rr = b * LSEG + l; }
  else if (l < 2 * LSEG)   { src = za; rr = b * LSEG + (l - LSEG); }
  else                     { src = zv; rr = b * LSEG + (l - 2 * LSEG); }
  xs[i] = src[(size_t)rr * D_MODEL + c];
}

// ---------------- pad W_x (136x512) -> (192x512), zeros below ----------------
__global__ void k_pad_wx(const float* __restrict__ wx, float* __restrict__ wxp) {
  int i = blockIdx.x * blockDim.x + threadIdx.x;
  if (i >= NXP * DI) return;
  int n = i / DI;
  wxp[i] = (n < DTR + 2 * NST) ? wx[i] : 0.f;
}

// ---------------- fp32 WMMA GEMM: Y(MxN,ldy) = X(MxK,ldx) * W(NxK)^T + bias ----
// block = 128 threads (4 waves); wave -> 32 rows x 64 cols (2x4 accum tiles).
// Software-pipelined: fragments for k0+4 are fetched before WMMAs for k0 issue.
// grid = (N/64, M/128)
__global__ __launch_bounds__(128)
void k_gemm_wmma(const float* __restrict__ X, int ldx,
                 const float* __restrict__ W,              // (N x K), ld = K
                 const float* __restrict__ bias,           // nullable, length N
                 float* __restrict__ Y, int ldy,
                 int N, int K) {
  const int lane = threadIdx.x & 31;
  const int wave = threadIdx.x >> 5;
  const int m0 = blockIdx.y * 128 + wave * 32;
  const int n0 = blockIdx.x * 64;
  const int lm = lane & 15;       // M (for A) / N (for B,C) within tile
  const int lh = lane >> 4;       // K half select

  const float* xrow0 = X + (size_t)(m0 + lm) * ldx + 2 * lh;
  const float* xrow1 = xrow0 + (size_t)16 * ldx;
  const float* wrow  = W + (size_t)(n0 + lm) * K + 2 * lh;
  const size_t ws1 = (size_t)16 * K;
  const size_t ws2 = 2 * ws1, ws3 = 3 * ws1;

  v8f c00 = {}, c01 = {}, c02 = {}, c03 = {};
  v8f c10 = {}, c11 = {}, c12 = {}, c13 = {};

  // prologue: fragments for k0 = 0
  v2f a0, a1, b0, b1, b2, b3;
  a0.x = xrow0[0];       a0.y = xrow0[1];
  a1.x = xrow1[0];       a1.y = xrow1[1];
  b0.x = wrow[0];        b0.y = wrow[1];
  b1.x = wrow[ws1];      b1.y = wrow[ws1 + 1];
  b2.x = wrow[ws2];      b2.y = wrow[ws2 + 1];
  b3.x = wrow[ws3];      b3.y = wrow[ws3 + 1];

  for (int k0 = 0; k0 < K; k0 += 4) {
    const int kn = (k0 + 4 < K) ? (k0 + 4) : k0;   // last iter: harmless reload
    v2f na0, na1, nb0, nb1, nb2, nb3;
    na0.x = xrow0[kn];       na0.y = xrow0[kn + 1];
    na1.x = xrow1[kn];       na1.y = xrow1[kn + 1];
    nb0.x = wrow[kn];        nb0.y = wrow[kn + 1];
    nb1.x = wrow[ws1 + kn];  nb1.y = wrow[ws1 + kn + 1];
    nb2.x = wrow[ws2 + kn];  nb2.y = wrow[ws2 + kn + 1];
    nb3.x = wrow[ws3 + kn];  nb3.y = wrow[ws3 + kn + 1];

    c00 = __builtin_amdgcn_wmma_f32_16x16x4_f32(false, a0, false, b0, (short)0, c00, false, false);
    c01 = __builtin_amdgcn_wmma_f32_16x16x4_f32(false, a0, false, b1, (short)0, c01, false, false);
    c02 = __builtin_amdgcn_wmma_f32_16x16x4_f32(false, a0, false, b2, (short)0, c02, false, false);
    c03 = __builtin_amdgcn_wmma_f32_16x16x4_f32(false, a0, false, b3, (short)0, c03, false, false);
    c10 = __builtin_amdgcn_wmma_f32_16x16x4_f32(false, a1, false, b0, (short)0, c10, false, false);
    c11 = __builtin_amdgcn_wmma_f32_16x16x4_f32(false, a1, false, b1, (short)0, c11, false, false);
    c12 = __builtin_amdgcn_wmma_f32_16x16x4_f32(false, a1, false, b2, (short)0, c12, false, false);
    c13 = __builtin_amdgcn_wmma_f32_16x16x4_f32(false, a1, false, b3, (short)0, c13, false, false);

    a0 = na0; a1 = na1; b0 = nb0; b1 = nb1; b2 = nb2; b3 = nb3;
  }

#define STORE_TILE(ACC, RT, TT)                                           \
  do {                                                                    \
    int col = n0 + 16 * (TT) + lm;                                        \
    float bv = bias ? bias[col] : 0.f;                                    \
    _Pragma("unroll")                                                     \
    for (int v = 0; v < 8; ++v)                                           \
      Y[(size_t)(m0 + 16 * (RT) + 8 * lh + v) * ldy + col] = ACC[v] + bv; \
  } while (0)
  STORE_TILE(c00, 0, 0);
  STORE_TILE(c01, 0, 1);
  STORE_TILE(c02, 0, 2);
  STORE_TILE(c03, 0, 3);
  STORE_TILE(c10, 1, 0);
  STORE_TILE(c11, 1, 1);
  STORE_TILE(c12, 1, 2);
  STORE_TILE(c13, 1, 3);
#undef STORE_TILE
}

// ---------------- causal depthwise conv (k=4) + SiLU ----------------
// mode 0: temporal seqs (len 1536, row stride 1); mode 1: modal (len 3, stride 512)
// dir +1: fwd taps x[p-3+k]; dir -1: flip-conv-flip == taps x[p+3-k]
__global__ void k_conv(const float* __restrict__ xz, const float* __restrict__ cw,
                       const float* __restrict__ cb, float* __restrict__ xc,
                       int mode, int dir) {
  int i = blockIdx.x * blockDim.x + threadIdx.x;
  if (i >= TTOK * DI) return;
  int r = i / DI, d = i % DI;
  int p, slen, rstep;
  if (mode == 0) { p = r % LTOT;          slen = LTOT; rstep = 1;    }
  else           { p = (r % LTOT) / LSEG; slen = 3;    rstep = LSEG; }
  float s = cb[d];
#pragma unroll
  for (int k = 0; k < 4; ++k) {
    int o = dir * (k - 3);
    int pp = p + o;
    if (pp >= 0 && pp < slen)
      s += cw[k * DI + d] * xz[(size_t)(r + o * rstep) * (2 * DI) + d];
  }
  xc[i] = s / (1.f + __expf(-s));   // silu
}

// ---------------- delta = softplus(dt @ W_dt^T + b_dt) ----------------
__global__ void k_delta(const float* __restrict__ xdbl, const float* __restrict__ wdt,
                        const float* __restrict__ bdt, float* __restrict__ delta) {
  int i = blockIdx.x * blockDim.x + threadIdx.x;
  if (i >= TTOK * DI) return;
  int r = i / DI, d = i % DI;
  float s = bdt[d];
#pragma unroll
  for (int k = 0; k < DTR; ++k)
    s += xdbl[(size_t)r * NXP + k] * wdt[d * DTR + k];
  delta[i] = (s > 20.f) ? s : log1pf(__expf(s));
}

// ---------------- selective scan + gating epilogue ----------------
// State-split version: each thread owns QSP=8 states of one channel.
// Thread index g = blockIdx.y*128 + tx; channel d = g>>3, octet q = g&7.
// The 8 octet lanes of a channel are consecutive lanes of one wave, so the
// per-step y = sum_n h[n]*C[n] finishes with 3 shfl_xor reductions.
// grid = (nseq, DI*QSP/128 = 32); block = 128 threads.
// xc holds conv output on entry; lane q==0 overwrites it with gated y.
__global__ __launch_bounds__(128)
void k_scan(const float* __restrict__ delta, const float* __restrict__ xdbl,
            const float* __restrict__ alog,  const float* __restrict__ dparm,
            const float* __restrict__ xz,    float* __restrict__ xc,
            int mode, int dir) {
  __shared__ float Bs[NST], Cs[NST];
  const int s = blockIdx.x;
  const int g = blockIdx.y * 128 + threadIdx.x;
  const int d = g >> 3;          // channel
  const int q = g & 7;           // state octet
  const int nb = q * QSP;        // first state index of this octet
  int base, rstep, slen;
  if (mode == 0) { base = s * LTOT;                       rstep = 1;    slen = LTOT; }
  else           { base = (s >> 9) * LTOT + (s & 511);    rstep = LSEG; slen = 3;    }

  float A[QSP], h[QSP];
#pragma unroll
  for (int i = 0; i < QSP; ++i) {
    A[i] = -__expf(alog[(size_t)d * NST + nb + i]);
    h[i] = 0.f;
  }
  const float Dp = dparm[d];

  for (int t = 0; t < slen; ++t) {
    const int p = (dir > 0) ? t : (slen - 1 - t);
    const size_t r = (size_t)base + (size_t)p * rstep;
    if (threadIdx.x < NST)            Bs[threadIdx.x]       = xdbl[r * NXP + DTR + threadIdx.x];
    else if (threadIdx.x < 2 * NST)   Cs[threadIdx.x - NST] = xdbl[r * NXP + DTR + NST + (threadIdx.x - NST)];
    __syncthreads();

    const float dl = delta[r * DI + d];
    const float xv = xc[r * DI + d];
    const float dx = dl * xv;
    float y = 0.f;
#pragma unroll
    for (int i = 0; i < QSP; ++i) {
      float w = __expf(dl * A[i]);
      h[i] = w * h[i] + dx * Bs[nb + i];
      y += h[i] * Cs[nb + i];
    }
    // reduce partial y over the 8 octet lanes of this channel
    y += __shfl_xor(y, 1, 32);
    y += __shfl_xor(y, 2, 32);
    y += __shfl_xor(y, 4, 32);
    if (q == 0) {
      const float zv = xz[r * (2 * DI) + DI + d];
      xc[r * DI + d] = (y + xv * Dp) * (zv / (1.f + __expf(-zv)));
    }
    __syncthreads();
  }
}

// ---------------- residual + LayerNorm (wave per row) ----------------
__global__ __launch_bounds__(32)
void k_ln(const float* __restrict__ hf, const float* __restrict__ xs,
          const float* __restrict__ g, const float* __restrict__ b,
          float* __restrict__ out) {
  const int r = blockIdx.x;
  const int lane = threadIdx.x;
  float v[4];
  float sum = 0.f;
#pragma unroll
  for (int j = 0; j < 4; ++j) {
    int c = lane + 32 * j;
    v[j] = hf[(size_t)r * D_MODEL + c] + xs[(size_t)r * D_MODEL + c];
    sum += v[j];
  }
#pragma unroll
  for (int o = 16; o > 0; o >>= 1) sum += __shfl_xor(sum, o, 32);
  const float mu = sum * (1.f / D_MODEL);
  float var = 0.f;
#pragma unroll
  for (int j = 0; j < 4; ++j) { float t = v[j] - mu; var += t * t; }
#pragma unroll
  for (int o = 16; o > 0; o >>= 1) var += __shfl_xor(var, o, 32);
  var *= (1.f / D_MODEL);
  const float rs = rsqrtf(var + 1e-5f);
#pragma unroll
  for (int j = 0; j < 4; ++j) {
    int c = lane + 32 * j;
    out[(size_t)r * D_MODEL + c] = (v[j] - mu) * rs * g[c] + b[c];
  }
}

// ---------------- host launcher ----------------
extern "C" void kernel_launch(void* const* d_in, const int* in_sizes, int n_in,
                              void* d_out, int out_size, void* d_ws, size_t ws_size,
                              hipStream_t stream) {
  (void)in_sizes; (void)n_in; (void)out_size; (void)ws_size;

  const float* zt = (const float*)d_in[0];
  const float* za = (const float*)d_in[1];
  const float* zv = (const float*)d_in[2];

  struct MP { const float *W_in, *conv_w, *conv_b, *W_x, *W_dt, *b_dt, *A_log, *Dp, *W_out; };
  struct LP { MP m[4]; const float *W_proj, *b_proj, *ln_g, *ln_b; };
  LP P[3];
  int idx = 3;
  for (int L = 0; L < 3; ++L) {
    for (int j = 0; j < 4; ++j) {
      MP& mp = P[L].m[j];
      mp.W_in   = (const float*)d_in[idx++];
      mp.conv_w = (const float*)d_in[idx++];
      mp.conv_b = (const float*)d_in[idx++];
      mp.W_x    = (const float*)d_in[idx++];
      mp.W_dt   = (const float*)d_in[idx++];
      mp.b_dt   = (const float*)d_in[idx++];
      mp.A_log  = (const float*)d_in[idx++];
      mp.Dp     = (const float*)d_in[idx++];
      mp.W_out  = (const float*)d_in[idx++];
    }
    P[L].W_proj = (const float*)d_in[idx++];
    P[L].b_proj = (const float*)d_in[idx++];
    P[L].ln_g   = (const float*)d_in[idx++];
    P[L].ln_b   = (const float*)d_in[idx++];
  }

  // workspace carve-up (floats); total ~38.6M floats ~ 155 MB
  float* ws   = (float*)d_ws;
  const size_t T = TTOK;
  float* xA   = ws;                          // T*128
  float* xB   = xA   + T * 128;              // T*128
  float* xz   = xB   + T * 128;              // T*1024
  float* xc   = xz   + T * 1024;             // T*512
  float* xdbl = xc   + T * 512;              // T*192
  float* dl   = xdbl + T * NXP;              // T*512
  float* wxp  = dl   + T * 512;              // 192*512
  float* hcat = wxp  + (size_t)NXP * DI;     // T*512
  float* hfus = hcat + T * 512;              // T*128

  k_concat<<<(int)((T * 128 + 255) / 256), 256, 0, stream>>>(zt, za, zv, xA);

  float* cur = xA;
  float* nxt = xB;
  const int MB = (int)(T / 128);  // 96 row-tiles of 128

  for (int L = 0; L < 3; ++L) {
    for (int j = 0; j < 4; ++j) {
      const MP& mp = P[L].m[j];
      const int mode = (j < 2) ? 0 : 1;
      const int dir  = (j % 2 == 0) ? 1 : -1;

      // in_proj: xz = cur @ W_in^T   (N=1024, K=128)
      k_gemm_wmma<<<dim3(1024 / 64, MB), 128, 0, stream>>>(cur, D_MODEL, mp.W_in, nullptr,
                                                           xz, 2 * DI, 2 * DI, D_MODEL);
      // depthwise conv + silu
      k_conv<<<(int)((T * DI + 255) / 256), 256, 0, stream>>>(xz, mp.conv_w, mp.conv_b, xc, mode, dir);
      // x_dbl = xc @ W_x^T (padded N=192, K=512)
      k_pad_wx<<<(NXP * DI + 255) / 256, 256, 0, stream>>>(mp.W_x, wxp);
      k_gemm_wmma<<<dim3(NXP / 64, MB), 128, 0, stream>>>(xc, DI, wxp, nullptr,
                                                          xdbl, NXP, NXP, DI);
      // delta = softplus(dt @ W_dt^T + b_dt)
      k_delta<<<(int)((T * DI + 255) / 256), 256, 0, stream>>>(xdbl, mp.W_dt, mp.b_dt, dl);
      // selective scan + gating (in-place into xc); 8-way state split per channel
      const int nseq = (mode == 0) ? BATCH : (BATCH * LSEG);
      k_scan<<<dim3(nseq, (DI * QSP) / 128), 128, 0, stream>>>(dl, xdbl, mp.A_log, mp.Dp,
                                                               xz, xc, mode, dir);
      // out_proj into h_cat columns [128j, 128j+128)
      k_gemm_wmma<<<dim3(128 / 64, MB), 128, 0, stream>>>(xc, DI, mp.W_out, nullptr,
                                                          hcat + 128 * j, 512, 128, DI);
    }
    // h_fused = h_cat @ W_proj^T + b_proj
    k_gemm_wmma<<<dim3(128 / 64, MB), 128, 0, stream>>>(hcat, 512, P[L].W_proj, P[L].b_proj,
                                                        hfus, D_MODEL, D_MODEL, 512);
    // x = LN(h_fused + x)
    float* outp = (L == 2) ? (float*)d_out : nxt;
    k_ln<<<(int)T, 32, 0, stream>>>(hfus, cur, P[L].ln_g, P[L].ln_b, outp);
    if (L < 2) { float* t = cur; cur = nxt; nxt = t; }
  }
}